// CTPNLoss_14319420964957
// MI455X (gfx1250) — compile-verified
//
#include <hip/hip_runtime.h>
#include <hip/hip_bf16.h>

// CTPN loss for MI455X (gfx1250, wave32).
// Latency-bound gather kernel: 1 wave, each lane owns 2 pos + 2 neg anchors,
// issues all gathers with max memory-level parallelism, then reduces the four
// scalar sums with V_WMMA_F32_16X16X4_F32 (A = partials, B = ones) in FP32.

typedef __attribute__((ext_vector_type(2))) float v2f;
typedef __attribute__((ext_vector_type(8))) float v8f;

#define H_DIM 512
#define W_DIM 1024
#define K_DIM 10
#define HW (H_DIM * W_DIM)

__device__ __forceinline__ float smooth_l1(float d) {
    float a = fabsf(d);
    return (a < 1.0f) ? (0.5f * a * a) : (a - 0.5f);
}

// softplus(d) = max(d,0) + log1p(exp(-|d|)) == logsumexp(0,d)
__device__ __forceinline__ float softplus(float d) {
    return fmaxf(d, 0.0f) + log1pf(expf(-fabsf(d)));
}

// Full 32-lane FP32 sum via one WMMA:
//   A[16x4]: this lane's value in A-vgpr0 slot, 0 in A-vgpr1 slot.
//   B[4x16]: all ones (layout-independent).
//   D[m][n] = rowsum_m(A)  for every column n.
// Lane l sums its 8 D VGPRs -> covers M=0..7 (l<16) or M=8..15 (l>=16) of one
// column; adding lane 0 and lane 16 gives the sum of all 64 A slots = sum of
// the 32 per-lane values. Requires EXEC == all ones (uniform flow here).
__device__ __forceinline__ float wmma_sum32(float x) {
    v2f a; a.x = x;    a.y = 0.0f;
    v2f b; b.x = 1.0f; b.y = 1.0f;
    v8f c = {};
    v8f d = __builtin_amdgcn_wmma_f32_16x16x4_f32(
        /*neg_a=*/false, a, /*neg_b=*/false, b,
        /*c_mod=*/(short)0, c, /*reuse_a=*/false, /*reuse_b=*/false);
    float s = d[0] + d[1] + d[2] + d[3] + d[4] + d[5] + d[6] + d[7];
    return __shfl(s, 0, 32) + __shfl(s, 16, 32);
}

__global__ void __launch_bounds__(32)
ctpn_loss_kernel(const float* __restrict__ scores,
                 const float* __restrict__ vcoords,
                 const float* __restrict__ sides,
                 const int* __restrict__ pos_y, const int* __restrict__ pos_x,
                 const int* __restrict__ pos_z,
                 const int* __restrict__ neg_y, const int* __restrict__ neg_x,
                 const int* __restrict__ neg_z,
                 const float* __restrict__ v_targets,
                 const int* __restrict__ side_mask,
                 const float* __restrict__ side_targets,
                 float* __restrict__ out) {
    const int lane = threadIdx.x;  // 0..31, single wave

    float cls_local = 0.0f;   // CE terms (pos + neg)
    float v_local   = 0.0f;   // SmoothL1 vertical-coord terms
    float o_local   = 0.0f;   // masked SmoothL1 side terms
    float cnt_local = 0.0f;   // mask count

#pragma unroll
    for (int r = 0; r < 2; ++r) {
        const int i = lane + r * 32;  // anchor index 0..63

        // ---- positive anchor i ----
        const int y = pos_y[i], x = pos_x[i], z = pos_z[i];
        const int base = y * W_DIM + x;
        const float s0 = scores[(2 * z)     * HW + base];
        const float s1 = scores[(2 * z + 1) * HW + base];
        cls_local += softplus(s0 - s1);               // -log_softmax[1]

        const float vp0 = vcoords[(2 * z)     * HW + base];
        const float vp1 = vcoords[(2 * z + 1) * HW + base];
        v_local += smooth_l1(vp0 - v_targets[2 * i]) +
                   smooth_l1(vp1 - v_targets[2 * i + 1]);

        const float sp = sides[z * HW + base];
        const float m  = (float)side_mask[i];
        o_local   += smooth_l1(sp - side_targets[i]) * m;
        cnt_local += m;

        // ---- negative anchor i ----
        const int yn = neg_y[i], xn = neg_x[i], zn = neg_z[i];
        const int bn = yn * W_DIM + xn;
        const float t0 = scores[(2 * zn)     * HW + bn];
        const float t1 = scores[(2 * zn + 1) * HW + bn];
        cls_local += softplus(t1 - t0);               // -log_softmax[0]
    }

    // Cross-lane reductions (EXEC all-ones: uniform control flow up to here).
    const float cls_sum = wmma_sum32(cls_local);
    const float v_sum   = wmma_sum32(v_local);
    const float o_sum   = wmma_sum32(o_local);
    const float cnt_sum = wmma_sum32(cnt_local);

    if (lane == 0) {
        const float avg_cls = cls_sum * (1.0f / 128.0f);        // (64+64) terms
        const float avg_v   = v_sum   * (1.0f / 128.0f);        // 64 anchors x 2
        const float avg_o   = (cnt_sum > 0.5f) ? (o_sum / cnt_sum) : 0.0f;
        out[0] = avg_cls + 1.0f * avg_v + 2.0f * avg_o;          // loss
        out[1] = avg_cls;
        out[2] = avg_v;
        out[3] = avg_o;
    }
}

extern "C" void kernel_launch(void* const* d_in, const int* in_sizes, int n_in,
                              void* d_out, int out_size, void* d_ws, size_t ws_size,
                              hipStream_t stream) {
    (void)in_sizes; (void)n_in; (void)out_size; (void)d_ws; (void)ws_size;
    ctpn_loss_kernel<<<1, 32, 0, stream>>>(
        (const float*)d_in[0],   // scores
        (const float*)d_in[1],   // vcoords
        (const float*)d_in[2],   // sides
        (const int*)d_in[3],     // pos_y
        (const int*)d_in[4],     // pos_x
        (const int*)d_in[5],     // pos_z
        (const int*)d_in[6],     // neg_y
        (const int*)d_in[7],     // neg_x
        (const int*)d_in[8],     // neg_z
        (const float*)d_in[9],   // v_targets [64,2]
        (const int*)d_in[10],    // side_mask
        (const float*)d_in[11],  // side_targets
        (float*)d_out);          // {loss, avg_cls, avg_v, avg_o}
}